// VQCLayer_58737972740656
// MI455X (gfx1250) — compile-verified
//
#include <hip/hip_runtime.h>

typedef __attribute__((ext_vector_type(2))) float v2f;
typedef __attribute__((ext_vector_type(8))) float v8f;

#define NQ       14
#define NLAYERS  4
#define BATCH    2048
#define SIZE     16384      // 2^14 amplitudes
#define THREADS  256        // 8 wave32s

// dynamic LDS layout, in floats
#define OFF_S0R  0
#define OFF_S0I  16384
#define OFF_S1R  32768
#define OFF_S1I  49152
#define OFF_URE  65536      // 16x16 fused gate, real
#define OFF_UIM  65792      // imag
#define OFF_UIN  66048      // -imag (WMMA f32 has no A-negate modifier)
#define OFF_RED  66304      // 16 reduction cells
#define OFF_X    66320      // 16
#define OFF_W    66336      // 64
#define LDS_FLOATS 66400    // 265,600 B  (<= 320 KB/workgroup)

__global__ __launch_bounds__(THREADS)
void vqc14_kernel(const float* __restrict__ x, const float* __restrict__ w,
                  float* __restrict__ out)
{
    extern __shared__ float sm[];
    const int tid  = threadIdx.x;
    const int b    = blockIdx.x;
    const int lane = tid & 31;
    const int wv   = tid >> 5;

    float* xs = sm + OFF_X;
    float* ws = sm + OFF_W;
    if (tid < NQ)         xs[tid] = x[b * NQ + tid];
    if (tid < NLAYERS*NQ) ws[tid] = w[tid];

    float* curR = sm + OFF_S0R; float* curI = sm + OFF_S0I;
    float* altR = sm + OFF_S1R; float* altI = sm + OFF_S1I;
    for (int i = tid; i < SIZE; i += THREADS) {
        curR[i] = (i == 0) ? 1.0f : 0.0f;
        curI[i] = 0.0f;
    }
    __syncthreads();

    float* uR  = sm + OFF_URE;
    float* uI  = sm + OFF_UIM;
    float* uIN = sm + OFF_UIN;

    for (int l = 0; l < NLAYERS; ++l) {
        // ===== three fused 4-wire groups (wires 0-3, 4-7, 8-11) via WMMA =====
        for (int grp = 0; grp < 3; ++grp) {
            const int q0 = grp * 4;
            // ---- build fused complex 16x16 gate: U[g][h] = prod_i G_{q0+i}[g_i][h_i],
            //      G_q = RX(w_lq) * RY(x_q) ----
            {
                const int g = tid >> 4, h = tid & 15;
                float pr = 1.0f, pi = 0.0f;
#pragma unroll
                for (int i = 0; i < 4; ++i) {
                    const int q  = q0 + i;
                    const int gi = (g >> (3 - i)) & 1;
                    const int hi = (h >> (3 - i)) & 1;
                    float sa, ca, sw, cw;
                    __sincosf(0.5f * xs[q],          &sa, &ca);
                    __sincosf(0.5f * ws[l*NQ + q],   &sw, &cw);
                    float gr, gm;
                    if (gi == hi) { gr = cw * ca;                 gm = gi ? sw*sa : -sw*sa; }
                    else          { gr = gi ? cw*sa : -cw*sa;     gm = -sw * ca;            }
                    const float npr = pr*gr - pi*gm;
                    const float npi = pr*gm + pi*gr;
                    pr = npr; pi = npi;
                }
                uR[tid] = pr; uI[tid] = pi; uIN[tid] = -pi;
            }
            __syncthreads();

            const int rb = 10 - q0;       // log2 of stride of the group's LSB wire
            const int R  = 1 << rb;

            // ---- A fragments (16x4 K-chunks of U), ISA layout:
            //      lanes 0-15: M=lane, {K0,K1}; lanes 16-31: M=lane-16, {K2,K3} ----
            v2f aRf[4], aIf[4], aNf[4];
            {
                const int m    = lane & 15;
                const int koff = (lane >> 4) * 2;
#pragma unroll
                for (int k = 0; k < 4; ++k) {
                    const int r0 = m * 16 + 4 * k + koff;
                    v2f t;
                    t.x = uR[r0];  t.y = uR[r0 + 1];  aRf[k] = t;
                    t.x = uI[r0];  t.y = uI[r0 + 1];  aIf[k] = t;
                    t.x = uIN[r0]; t.y = uIN[r0 + 1]; aNf[k] = t;
                }
            }

            // ---- 64 column-tiles of 16; 8 per wave; state viewed as 16x1024 ----
            for (int t = wv * 8; t < wv * 8 + 8; ++t) {
                const int j    = t * 16 + (lane & 15);                    // column index
                const int base = ((j >> rb) << (rb + 4)) + (j & (R - 1)); // addr(h=0, col j)
                const int hoff = (lane >> 4) * 2;

                v2f bRf[4], bIf[4];
#pragma unroll
                for (int k = 0; k < 4; ++k) {
                    const int a0 = base + ((4 * k + hoff) << rb);
                    const int a1 = a0 + (1 << rb);
                    v2f tt;
                    tt.x = curR[a0]; tt.y = curR[a1]; bRf[k] = tt;
                    tt.x = curI[a0]; tt.y = curI[a1]; bIf[k] = tt;
                }

                v8f accR = {}; v8f accI = {};
#pragma unroll
                for (int k = 0; k < 4; ++k)   // out_re += U_re * S_re
                    accR = __builtin_amdgcn_wmma_f32_16x16x4_f32(false, aRf[k], false, bRf[k], (short)0, accR, false, false);
#pragma unroll
                for (int k = 0; k < 4; ++k)   // out_re += (-U_im) * S_im
                    accR = __builtin_amdgcn_wmma_f32_16x16x4_f32(false, aNf[k], false, bIf[k], (short)0, accR, false, false);
#pragma unroll
                for (int k = 0; k < 4; ++k)   // out_im += U_re * S_im
                    accI = __builtin_amdgcn_wmma_f32_16x16x4_f32(false, aRf[k], false, bIf[k], (short)0, accI, false, false);
#pragma unroll
                for (int k = 0; k < 4; ++k)   // out_im += U_im * S_re
                    accI = __builtin_amdgcn_wmma_f32_16x16x4_f32(false, aIf[k], false, bRf[k], (short)0, accI, false, false);

                const int mo = (lane >> 4) * 8;   // C/D layout: VGPR v -> rows v / v+8
#pragma unroll
                for (int v = 0; v < 8; ++v) {
                    const int addr = base + ((v + mo) << rb);
                    curR[addr] = accR[v];
                    curI[addr] = accI[v];
                }
            }
            __syncthreads();
        }

        // ===== wires 12,13 (4x4 complex gate) fused with the CNOT-ring permutation =====
        if (tid < 16) {
            const int g = tid >> 2, h = tid & 3;
            float pr = 1.0f, pi = 0.0f;
#pragma unroll
            for (int i = 0; i < 2; ++i) {
                const int q  = 12 + i;
                const int gi = (g >> (1 - i)) & 1;
                const int hi = (h >> (1 - i)) & 1;
                float sa, ca, sw, cw;
                __sincosf(0.5f * xs[q],        &sa, &ca);
                __sincosf(0.5f * ws[l*NQ + q], &sw, &cw);
                float gr, gm;
                if (gi == hi) { gr = cw * ca;             gm = gi ? sw*sa : -sw*sa; }
                else          { gr = gi ? cw*sa : -cw*sa; gm = -sw * ca;            }
                const float npr = pr*gr - pi*gm;
                const float npi = pr*gm + pi*gr;
                pr = npr; pi = npi;
            }
            uR[tid] = pr; uI[tid] = pi;
        }
        __syncthreads();

        for (int a = tid; a < SIZE / 4; a += THREADS) {
            float ir[4], ii[4];
#pragma unroll
            for (int h = 0; h < 4; ++h) { ir[h] = curR[a*4 + h]; ii[h] = curI[a*4 + h]; }
#pragma unroll
            for (int g = 0; g < 4; ++g) {
                float orr = 0.0f, oii = 0.0f;
#pragma unroll
                for (int h = 0; h < 4; ++h) {
                    const float gr = uR[g*4 + h], gm = uI[g*4 + h];
                    orr += gr*ir[h] - gm*ii[h];
                    oii += gr*ii[h] + gm*ir[h];
                }
                // CNOT ring (q -> q+1 mod 14) as GF(2) prefix-XOR permutation of index bits
                const unsigned s = (unsigned)(a * 4 + g);
                unsigned y = s;
                y ^= y >> 1; y ^= y >> 2; y ^= y >> 4; y ^= y >> 8;
                const unsigned d = (y & 0x1FFFu) | (((y ^ (s >> 13)) & 1u) << 13);
                altR[d] = orr; altI[d] = oii;
            }
        }
        __syncthreads();
        { float* t = curR; curR = altR; altR = t; }
        { float* t = curI; curI = altI; altI = t; }
    }

    // ===== <Z_q> = total - 2 * sum(prob | bit_q == 1) =====
    float* red = sm + OFF_RED;
    if (tid < 16) red[tid] = 0.0f;
    __syncthreads();

    float tot = 0.0f, s1[NQ];
#pragma unroll
    for (int q = 0; q < NQ; ++q) s1[q] = 0.0f;
    for (int i = tid; i < SIZE; i += THREADS) {
        const float p = curR[i]*curR[i] + curI[i]*curI[i];
        tot += p;
#pragma unroll
        for (int q = 0; q < NQ; ++q)
            s1[q] += p * (float)((i >> (13 - q)) & 1);
    }
#pragma unroll
    for (int off = 16; off > 0; off >>= 1) {
        tot += __shfl_down(tot, off, 32);
#pragma unroll
        for (int q = 0; q < NQ; ++q) s1[q] += __shfl_down(s1[q], off, 32);
    }
    if (lane == 0) {
        atomicAdd(&red[14], tot);
#pragma unroll
        for (int q = 0; q < NQ; ++q) atomicAdd(&red[q], s1[q]);
    }
    __syncthreads();
    if (tid < NQ) out[b * NQ + tid] = red[14] - 2.0f * red[tid];
}

extern "C" void kernel_launch(void* const* d_in, const int* in_sizes, int n_in,
                              void* d_out, int out_size, void* d_ws, size_t ws_size,
                              hipStream_t stream) {
    (void)in_sizes; (void)n_in; (void)out_size; (void)d_ws; (void)ws_size;
    const float* x = (const float*)d_in[0];   // [2048,14]
    const float* w = (const float*)d_in[1];   // [4,14]
    float* out = (float*)d_out;               // [2048,14]
    const size_t shmem = (size_t)LDS_FLOATS * sizeof(float);
    (void)hipFuncSetAttribute(reinterpret_cast<const void*>(&vqc14_kernel),
                              hipFuncAttributeMaxDynamicSharedMemorySize, (int)shmem);
    vqc14_kernel<<<BATCH, THREADS, shmem, stream>>>(x, w, out);
}